// MLPQuadrupedProjectionFilter_27728308863919
// MI455X (gfx1250) — compile-verified
//
#include <hip/hip_runtime.h>
#include <math.h>

// ---------------------------------------------------------------------------
// MLPQuadrupedProjectionFilter for MI455X (gfx1250, wave32, WMMA).
// fp32 end-to-end; all GEMMs via V_WMMA_F32_16X16X4_F32 with double-buffered
// GLOBAL_LOAD_ASYNC_TO_LDS_B32 tile staging (ASYNCcnt pipeline).
// ---------------------------------------------------------------------------

#define B_      4096
#define NVAR    120
#define NC      240
#define HID     1024
#define RIN     1080          // 3*(NC+NVAR)
#define G3      3072          // 3*HID
#define MAXITER 8
#define RHO     1.0f
#define REG     1e-4f

typedef float v2f __attribute__((ext_vector_type(2)));
typedef float v8f __attribute__((ext_vector_type(8)));

// ===========================================================================
// Tiled fp32 WMMA GEMM.
//   TRANSB=0 : D[M,N] = A[M,K] * B[K,N]      (B row-major K x N)
//   TRANSB=1 : D[M,N] = A[M,K] * W[N,K]^T    (W row-major N x K, torch layout)
// Block tile 64(M) x 128(N); K staged in 32-chunks, double-buffered in LDS via
// async-to-LDS loads (24 b32 ops per wave per chunk -> exact counted waits).
// Final partial-K chunk uses synchronous zero-padded staging (K-padding must
// be exactly zero; N-edge rows are clamped -> garbage only in discarded cols).
// M must be a multiple of 64 (always B_=4096 here); N,K arbitrary.
// ===========================================================================
#define TILE_M 64
#define TILE_N 128
#define TILE_K 32
#define LDT    34   // padded LDS row stride (even -> 8B-aligned v2f reads)

template <bool TRANSB>
__global__ __launch_bounds__(256) void gemm_f32_wmma(
    const float* __restrict__ A, const float* __restrict__ Bm,
    float* __restrict__ D, int M, int N, int K) {
  __shared__ __align__(16) float As[2][TILE_M * LDT];
  __shared__ __align__(16) float Bs[2][TILE_N * LDT];

  const int tid   = threadIdx.x;
  const int lane  = tid & 31;
  const int wave  = tid >> 5;
  const int m0    = blockIdx.y * TILE_M;
  const int n0    = blockIdx.x * TILE_N;
  const int waveM = (wave >> 2) * 32;   // 2 wave-rows
  const int waveN = (wave & 3) * 32;    // 4 wave-cols
  const int mr    = lane & 15;
  const int hi    = lane >> 4;          // k-half selector (ISA A 16x4 layout)

  v8f acc[2][2];
  for (int a = 0; a < 2; ++a)
    for (int b = 0; b < 2; ++b)
      for (int v = 0; v < 8; ++v) acc[a][b][v] = 0.0f;

  // ---- async staging of one full 32-wide K chunk into LDS buffer `buf` ----
  // Exactly 8 (A) + 16 (B) = 24 async instructions per wave, unconditionally.
  auto stage_async = [&](int k0, int buf) {
    for (int t = 0; t < (TILE_M * TILE_K) / 256; ++t) {
      int idx = tid + t * 256;
      int mm = idx >> 5, kk = idx & 31;
      unsigned lds = (unsigned)(uintptr_t)&As[buf][mm * LDT + kk];
      unsigned long long ga =
          (unsigned long long)(uintptr_t)(A + (size_t)(m0 + mm) * K + (k0 + kk));
      asm volatile("global_load_async_to_lds_b32 %0, %1, off"
                   :: "v"(lds), "v"(ga) : "memory");
    }
    for (int t = 0; t < (TILE_N * TILE_K) / 256; ++t) {
      int idx = tid + t * 256;
      int nn, kk;
      if (TRANSB) { nn = idx >> 5; kk = idx & 31; }
      else        { kk = idx >> 7; nn = idx & 127; }
      int gn = n0 + nn;
      if (gn > N - 1) gn = N - 1;   // clamp: dup row, cols >= N are discarded
      size_t goff = TRANSB ? ((size_t)gn * K + (k0 + kk))
                           : ((size_t)(k0 + kk) * N + gn);
      unsigned lds = (unsigned)(uintptr_t)&Bs[buf][nn * LDT + kk];
      unsigned long long ga = (unsigned long long)(uintptr_t)(Bm + goff);
      asm volatile("global_load_async_to_lds_b32 %0, %1, off"
                   :: "v"(lds), "v"(ga) : "memory");
    }
  };

  // ---- synchronous zero-padded staging (K-edge chunk only) ----
  auto stage_sync = [&](int k0, int buf) {
    for (int t = 0; t < (TILE_M * TILE_K) / 256; ++t) {
      int idx = tid + t * 256;
      int mm = idx >> 5, kk = idx & 31;
      int gk = k0 + kk;
      float v = 0.0f;
      if (gk < K) v = A[(size_t)(m0 + mm) * K + gk];
      As[buf][mm * LDT + kk] = v;
    }
    for (int t = 0; t < (TILE_N * TILE_K) / 256; ++t) {
      int idx = tid + t * 256;
      float v = 0.0f;
      if (TRANSB) {
        int nn = idx >> 5, kk = idx & 31;
        int gn = n0 + nn, gk = k0 + kk;
        if (gn < N && gk < K) v = Bm[(size_t)gn * K + gk];
        Bs[buf][nn * LDT + kk] = v;
      } else {
        int kk = idx >> 7, nn = idx & 127;
        int gn = n0 + nn, gk = k0 + kk;
        if (gn < N && gk < K) v = Bm[(size_t)gk * N + gn];
        Bs[buf][nn * LDT + kk] = v;
      }
    }
  };

  // ---- 8 k-steps x 4 WMMA on the staged chunk ----
  auto compute_tile = [&](int buf) {
#pragma unroll
    for (int kc = 0; kc < TILE_K; kc += 4) {
      // A frag 16x4: lane<16 -> k={0,1}; lane>=16 -> k={2,3}  (ISA 7.12.2)
      v2f a0 = *(const v2f*)&As[buf][(waveM + mr) * LDT + kc + 2 * hi];
      v2f a1 = *(const v2f*)&As[buf][(waveM + 16 + mr) * LDT + kc + 2 * hi];
      // B frag 4x16, mirrored layout; Bs is [n][k] so addressing is identical
      v2f b0 = *(const v2f*)&Bs[buf][(waveN + mr) * LDT + kc + 2 * hi];
      v2f b1 = *(const v2f*)&Bs[buf][(waveN + 16 + mr) * LDT + kc + 2 * hi];
      acc[0][0] = __builtin_amdgcn_wmma_f32_16x16x4_f32(false, a0, false, b0, (short)0, acc[0][0], false, false);
      acc[0][1] = __builtin_amdgcn_wmma_f32_16x16x4_f32(false, a0, false, b1, (short)0, acc[0][1], false, false);
      acc[1][0] = __builtin_amdgcn_wmma_f32_16x16x4_f32(false, a1, false, b0, (short)0, acc[1][0], false, false);
      acc[1][1] = __builtin_amdgcn_wmma_f32_16x16x4_f32(false, a1, false, b1, (short)0, acc[1][1], false, false);
    }
  };

  const int kFull = (K / TILE_K) * TILE_K;
  int cur = 0;
  if (kFull > 0) stage_async(0, 0);
  for (int k0 = 0; k0 < kFull; k0 += TILE_K) {
    const bool pf = (k0 + TILE_K < kFull);
    if (pf) {
      stage_async(k0 + TILE_K, cur ^ 1);            // prefetch next chunk
      asm volatile("s_wait_asynccnt 0x18" ::: "memory");  // only 24 newest left
    } else {
      asm volatile("s_wait_asynccnt 0x0" ::: "memory");
    }
    __syncthreads();
    compute_tile(cur);
    __syncthreads();
    cur ^= 1;
  }
  if (kFull < K) {                                  // zero-padded K edge
    stage_sync(kFull, cur);
    __syncthreads();
    compute_tile(cur);
  }

  // ---- store: D[m][n], C/D layout m = v + 8*hi, n = mr ----
  for (int tm = 0; tm < 2; ++tm)
    for (int tn = 0; tn < 2; ++tn)
      for (int v = 0; v < 8; ++v) {
        int m = m0 + waveM + 16 * tm + v + 8 * hi;
        int n = n0 + waveN + 16 * tn + mr;
        if (n < N) D[(size_t)m * N + n] = acc[tm][tn][v];
      }
}

// ===========================================================================
// One-time setup kernels
// ===========================================================================

// Build augmented systems: aug = [H + rho*C^T C + eps*I | I]  (120 x 240)
//                          aug1 = [H + eps*I | -g]            (120 x 121)
__global__ void k_build_aug(const float* __restrict__ H, const float* __restrict__ g,
                            const float* __restrict__ C, float* aug, float* aug1) {
  int idx = blockIdx.x * blockDim.x + threadIdx.x;
  if (idx < NVAR * NC) {
    int i = idx / NC, j = idx % NC;
    float v;
    if (j < NVAR) {
      float d = 0.0f;
      for (int k = 0; k < NC; ++k) d += C[k * NVAR + i] * C[k * NVAR + j];
      v = H[i * NVAR + j] + RHO * d + ((i == j) ? REG : 0.0f);
    } else {
      v = ((j - NVAR) == i) ? 1.0f : 0.0f;
    }
    aug[idx] = v;
  } else {
    int t = idx - NVAR * NC;
    if (t < NVAR * (NVAR + 1)) {
      int i = t / (NVAR + 1), j = t % (NVAR + 1);
      aug1[t] = (j < NVAR) ? (H[i * NVAR + j] + ((i == j) ? REG : 0.0f)) : -g[i];
    }
  }
}

// Gauss-Jordan on SPD augmented system, single wave32 block (one SIMD/L0,
// deterministic, no cross-CU coherence concerns). n pivots, width W.
__global__ void k_gauss_jordan(float* aug, int n, int W) {
  int lane = threadIdx.x;
  for (int p = 0; p < n; ++p) {
    float pivinv = 1.0f / aug[p * W + p];   // lockstep read before any write
    for (int c2 = lane; c2 < W; c2 += 32) aug[p * W + c2] *= pivinv;
    __threadfence_block();
    __syncthreads();
    for (int r = lane; r < n; r += 32) {
      if (r == p) continue;
      float f = aug[r * W + p];
      for (int c2 = 0; c2 < W; ++c2) aug[r * W + c2] -= f * aug[p * W + c2];
    }
    __threadfence_block();
    __syncthreads();
  }
}

__global__ void k_extract(const float* __restrict__ aug, const float* __restrict__ aug1,
                          float* qinv, float* xi0v) {
  int idx = blockIdx.x * blockDim.x + threadIdx.x;
  if (idx < NVAR * NVAR) {
    qinv[idx] = aug[(idx / NVAR) * NC + NVAR + (idx % NVAR)];
  } else if (idx < NVAR * NVAR + NVAR) {
    int i = idx - NVAR * NVAR;
    xi0v[i] = aug1[i * (NVAR + 1) + NVAR];
  }
}

// s0 vector (batch-independent): s0 = max(0, c - C @ xi0)
__global__ void k_s0(const float* __restrict__ xi0v, const float* __restrict__ C,
                     const float* __restrict__ c, float* s0v) {
  int j = blockIdx.x * blockDim.x + threadIdx.x;
  if (j < NC) {
    float a = 0.0f;
    for (int i = 0; i < NVAR; ++i) a += C[j * NVAR + i] * xi0v[i];
    s0v[j] = fmaxf(0.0f, c[j] - a);
  }
}

__global__ void k_init_state(const float* __restrict__ s0v, const float* __restrict__ lam0,
                             const float* __restrict__ h0, float* s, float* lam, float* h) {
  int idx = blockIdx.x * blockDim.x + threadIdx.x;
  const int n_s = B_ * NC, n_l = B_ * NVAR;
  if (idx < n_s) {
    s[idx] = s0v[idx % NC];
  } else if (idx < n_s + n_l) {
    lam[idx - n_s] = lam0[idx - n_s];
  } else if (idx < n_s + n_l + B_ * HID) {
    h[idx - n_s - n_l] = h0[idx - n_s - n_l];
  }
}

// ===========================================================================
// Per-iteration elementwise / reduction kernels
// ===========================================================================

__global__ void k_baug(const float* __restrict__ s, const float* __restrict__ c, float* t240) {
  int idx = blockIdx.x * blockDim.x + threadIdx.x;
  if (idx < B_ * NC) t240[idx] = c[idx % NC] - s[idx];
}

// rhs = -lincost = lam - g + rho * ((c-s)@C)
__global__ void k_rhs(const float* __restrict__ lam, const float* __restrict__ g,
                      const float* __restrict__ t120, float* rhs) {
  int idx = blockIdx.x * blockDim.x + threadIdx.x;
  if (idx < B_ * NVAR) rhs[idx] = lam[idx] - g[idx % NVAR] + RHO * t120[idx];
}

// s_new = max(0, c - Axi); res_vec = Axi - c + s_new  (res written in place of Axi)
__global__ void k_snew_res(const float* __restrict__ c, float* axi, float* snew) {
  int idx = blockIdx.x * blockDim.x + threadIdx.x;
  if (idx < B_ * NC) {
    float a = axi[idx], cc = c[idx % NC];
    float sn = fmaxf(0.0f, cc - a);
    snew[idx] = sn;
    axi[idx] = a - cc + sn;
  }
}

__global__ void k_lamnew(const float* __restrict__ lam, const float* __restrict__ t120, float* lamnew) {
  int idx = blockIdx.x * blockDim.x + threadIdx.x;
  if (idx < B_ * NVAR) lamnew[idx] = lam[idx] - RHO * t120[idx];
}

// dst[row] = || src[row,:] ||_2   (block per row, fixed-order LDS reduction)
__global__ void k_rownorm(const float* __restrict__ src, int cols, float* dst) {
  __shared__ float red[256];
  int row = blockIdx.x;
  float a = 0.0f;
  for (int c2 = threadIdx.x; c2 < cols; c2 += 256) {
    float v = src[(size_t)row * cols + c2];
    a += v * v;
  }
  red[threadIdx.x] = a;
  __syncthreads();
  for (int s2 = 128; s2 > 0; s2 >>= 1) {
    if (threadIdx.x < (unsigned)s2) red[threadIdx.x] += red[threadIdx.x + s2];
    __syncthreads();
  }
  if (threadIdx.x == 0) dst[row] = sqrtf(red[0]);
}

// dst[row] = A[row,:] . v
__global__ void k_rowdot(const float* __restrict__ A, const float* __restrict__ v,
                         int cols, float* dst) {
  __shared__ float red[256];
  int row = blockIdx.x;
  float a = 0.0f;
  for (int c2 = threadIdx.x; c2 < cols; c2 += 256) a += A[(size_t)row * cols + c2] * v[c2];
  red[threadIdx.x] = a;
  __syncthreads();
  for (int s2 = 128; s2 > 0; s2 >>= 1) {
    if (threadIdx.x < (unsigned)s2) red[threadIdx.x] += red[threadIdx.x + s2];
    __syncthreads();
  }
  if (threadIdx.x == 0) dst[row] = red[0];
}

// qpres[i] = || 0.5*(Y @ xi^T)[i,:] + xg ||_2 over the full 4096-wide row,
// fused (never materializes the [B,B] matrix). Block = 16 rows, thread owns
// (r, js) pair; deterministic fixed-order reduction.
__global__ __launch_bounds__(256) void k_qpnorm(const float* __restrict__ Y,
                                                const float* __restrict__ xi,
                                                const float* __restrict__ xg, float* dst) {
  __shared__ float Yt[16 * NVAR];
  __shared__ float Xt[16 * NVAR];
  __shared__ float Xgs[16];
  __shared__ float red[256];
  int r0 = blockIdx.x * 16;
  for (int idx = threadIdx.x; idx < 16 * NVAR; idx += 256)
    Yt[idx] = Y[(size_t)(r0 + idx / NVAR) * NVAR + (idx % NVAR)];
  int r = threadIdx.x >> 4, js = threadIdx.x & 15;
  float acc = 0.0f;
  for (int jt = 0; jt < B_ / 16; ++jt) {
    int j0 = jt * 16;
    __syncthreads();
    for (int idx = threadIdx.x; idx < 16 * NVAR; idx += 256)
      Xt[idx] = xi[(size_t)(j0 + idx / NVAR) * NVAR + (idx % NVAR)];
    if (threadIdx.x < 16) Xgs[threadIdx.x] = xg[j0 + threadIdx.x];
    __syncthreads();
    float d = 0.0f;
#pragma unroll 8
    for (int k = 0; k < NVAR; ++k) d += Yt[r * NVAR + k] * Xt[js * NVAR + k];
    float q = 0.5f * d + Xgs[js];
    acc += q * q;
  }
  red[threadIdx.x] = acc;
  __syncthreads();
  if (threadIdx.x < 16) {
    float ssum = 0.0f;
    for (int t = 0; t < 16; ++t) ssum += red[threadIdx.x * 16 + t];
    dst[r0 + threadIdx.x] = sqrtf(ssum);
  }
}

// r = concat([s, lam, s_new, lam_new, s_new - s, lam_new - lam])
__global__ void k_concat(const float* __restrict__ s, const float* __restrict__ lam,
                         const float* __restrict__ snew, const float* __restrict__ lamnew,
                         float* rb) {
  int idx = blockIdx.x * blockDim.x + threadIdx.x;
  if (idx >= B_ * RIN) return;
  int b = idx / RIN, j = idx % RIN;
  float v;
  if (j < NC)                 v = s[(size_t)b * NC + j];
  else if (j < NC + NVAR)     v = lam[(size_t)b * NVAR + (j - NC)];
  else if (j < 2 * NC + NVAR) v = snew[(size_t)b * NC + (j - NC - NVAR)];
  else if (j < 2 * NC + 2 * NVAR) v = lamnew[(size_t)b * NVAR + (j - 2 * NC - NVAR)];
  else if (j < 3 * NC + 2 * NVAR) {
    int t = j - 2 * NC - 2 * NVAR;
    v = snew[(size_t)b * NC + t] - s[(size_t)b * NC + t];
  } else {
    int t = j - 3 * NC - 2 * NVAR;
    v = lamnew[(size_t)b * NVAR + t] - lam[(size_t)b * NVAR + t];
  }
  rb[idx] = v;
}

// GRU gates; h updated in place (elementwise, old h read first)
__global__ void k_gates(const float* __restrict__ gi, const float* __restrict__ gh,
                        const float* __restrict__ bih, const float* __restrict__ bhh,
                        float* h) {
  int idx = blockIdx.x * blockDim.x + threadIdx.x;
  if (idx >= B_ * HID) return;
  int b = idx / HID, j = idx % HID;
  size_t base = (size_t)b * G3;
  float ir = gi[base + j] + bih[j];
  float iz = gi[base + HID + j] + bih[HID + j];
  float in_ = gi[base + 2 * HID + j] + bih[2 * HID + j];
  float hr = gh[base + j] + bhh[j];
  float hz = gh[base + HID + j] + bhh[HID + j];
  float hn = gh[base + 2 * HID + j] + bhh[2 * HID + j];
  float rr = 1.0f / (1.0f + expf(-(ir + hr)));
  float zz = 1.0f / (1.0f + expf(-(iz + hz)));
  float nn = tanhf(in_ + rr * hn);
  float ho = h[idx];
  h[idx] = (1.0f - zz) * nn + zz * ho;
}

// Final refinement + fixed-point residual norm; s, lam updated in place.
__global__ void k_final(const float* __restrict__ outb, const float* __restrict__ bout,
                        const float* __restrict__ snew, const float* __restrict__ lamnew,
                        float* s, float* lam, float* fixed_dst) {
  __shared__ float red[256];
  __shared__ float snorm;
  int b = blockIdx.x;
  float a = 0.0f;
  for (int j = threadIdx.x; j < NC; j += 256) {
    float o = outb[(size_t)b * (NC + NVAR) + j] + bout[j];
    float sf = fmaxf(0.0f, snew[(size_t)b * NC + j] + o);
    float d = sf - s[(size_t)b * NC + j];
    a += d * d;
    s[(size_t)b * NC + j] = sf;
  }
  red[threadIdx.x] = a;
  __syncthreads();
  for (int s2 = 128; s2 > 0; s2 >>= 1) {
    if (threadIdx.x < (unsigned)s2) red[threadIdx.x] += red[threadIdx.x + s2];
    __syncthreads();
  }
  if (threadIdx.x == 0) snorm = sqrtf(red[0]);
  __syncthreads();
  float a2 = 0.0f;
  for (int v = threadIdx.x; v < NVAR; v += 256) {
    float o = outb[(size_t)b * (NC + NVAR) + NC + v] + bout[NC + v];
    float lf = lamnew[(size_t)b * NVAR + v] + o;
    float d = lf - lam[(size_t)b * NVAR + v];
    a2 += d * d;
    lam[(size_t)b * NVAR + v] = lf;
  }
  red[threadIdx.x] = a2;
  __syncthreads();
  for (int s2 = 128; s2 > 0; s2 >>= 1) {
    if (threadIdx.x < (unsigned)s2) red[threadIdx.x] += red[threadIdx.x + s2];
    __syncthreads();
  }
  if (threadIdx.x == 0) fixed_dst[b] = snorm + sqrtf(red[0]);
}

// ===========================================================================
// Host launcher
// ===========================================================================
extern "C" void kernel_launch(void* const* d_in, const int* in_sizes, int n_in,
                              void* d_out, int out_size, void* d_ws, size_t ws_size,
                              hipStream_t stream) {
  (void)in_sizes; (void)n_in; (void)out_size;
  const float* lam0  = (const float*)d_in[0];
  // d_in[1] = s_init (multiplied by 0 in reference -> unused)
  const float* h0    = (const float*)d_in[2];
  const float* H     = (const float*)d_in[3];
  const float* g     = (const float*)d_in[4];
  const float* C     = (const float*)d_in[5];
  const float* c     = (const float*)d_in[6];
  const float* W_ih  = (const float*)d_in[7];
  const float* b_ih  = (const float*)d_in[8];
  const float* W_hh  = (const float*)d_in[9];
  const float* b_hh  = (const float*)d_in[10];
  const float* W_out = (const float*)d_in[11];
  const float* b_out = (const float*)d_in[12];

  float* out    = (float*)d_out;
  float* xi     = out;                                 // [B,NVAR]; last iter persists
  float* primal = out + (size_t)B_ * NVAR;             // [8,B]
  float* fixedp = primal + (size_t)MAXITER * B_;       // [8,B]
  float* qpres  = fixedp + (size_t)MAXITER * B_;       // [8,B]

  float* Wf = (float*)d_ws;
  size_t off = 0;
  auto alloc = [&](size_t n) {
    float* p = Wf + off;
    off += (n + 255) & ~(size_t)255;
    return p;
  };
  float* aug    = alloc(NVAR * NC);
  float* aug1   = alloc(NVAR * (NVAR + 1));
  float* qinv   = alloc(NVAR * NVAR);
  float* xi0v   = alloc(NVAR);
  float* s0v    = alloc(NC);
  float* xg     = alloc(B_);
  float* s      = alloc((size_t)B_ * NC);
  float* lam    = alloc((size_t)B_ * NVAR);
  float* h      = alloc((size_t)B_ * HID);
  float* t240   = alloc((size_t)B_ * NC);
  float* t120   = alloc((size_t)B_ * NVAR);
  float* rhs    = alloc((size_t)B_ * NVAR);
  float* axi    = alloc((size_t)B_ * NC);    // becomes res_vec in place
  float* snew   = alloc((size_t)B_ * NC);
  float* lamnew = alloc((size_t)B_ * NVAR);
  float* ybuf   = alloc((size_t)B_ * NVAR);
  float* rbuf   = alloc((size_t)B_ * RIN);
  float* gi     = alloc((size_t)B_ * G3);
  float* gh     = alloc((size_t)B_ * G3);
  float* outb   = rbuf;  // alias: rbuf dead after gi GEMM; out head [B,360]
  if (ws_size < off * sizeof(float)) return;  // insufficient scratch

  auto cdiv = [](int a, int b) { return (a + b - 1) / b; };

  // ---- one-time setup ----
  k_build_aug<<<cdiv(NVAR * NC + NVAR * (NVAR + 1), 256), 256, 0, stream>>>(H, g, C, aug, aug1);
  k_gauss_jordan<<<1, 32, 0, stream>>>(aug, NVAR, NC);         // Q_inv = (H+rho C^T C+eps I)^-1
  k_gauss_jordan<<<1, 32, 0, stream>>>(aug1, NVAR, NVAR + 1);  // xi0 = (H+eps I)^-1 (-g)
  k_extract<<<cdiv(NVAR * NVAR + NVAR, 256), 256, 0, stream>>>(aug, aug1, qinv, xi0v);
  k_s0<<<1, 256, 0, stream>>>(xi0v, C, c, s0v);
  k_init_state<<<cdiv(B_ * (NC + NVAR + HID), 256), 256, 0, stream>>>(s0v, lam0, h0, s, lam, h);

  // ---- ADMM + GRU iterations ----
  for (int it = 0; it < MAXITER; ++it) {
    k_baug<<<cdiv(B_ * NC, 256), 256, 0, stream>>>(s, c, t240);
    gemm_f32_wmma<false><<<dim3(cdiv(NVAR, TILE_N), B_ / TILE_M), 256, 0, stream>>>(
        t240, C, t120, B_, NVAR, NC);                           // (c-s) @ C
    k_rhs<<<cdiv(B_ * NVAR, 256), 256, 0, stream>>>(lam, g, t120, rhs);
    gemm_f32_wmma<false><<<dim3(cdiv(NVAR, TILE_N), B_ / TILE_M), 256, 0, stream>>>(
        rhs, qinv, xi, B_, NVAR, NVAR);                         // xi = rhs @ Qinv (sym)
    gemm_f32_wmma<true><<<dim3(cdiv(NC, TILE_N), B_ / TILE_M), 256, 0, stream>>>(
        xi, C, axi, B_, NC, NVAR);                              // Axi = xi @ C^T
    k_snew_res<<<cdiv(B_ * NC, 256), 256, 0, stream>>>(c, axi, snew);
    k_rownorm<<<B_, 256, 0, stream>>>(axi, NC, primal + (size_t)it * B_);
    gemm_f32_wmma<false><<<dim3(cdiv(NVAR, TILE_N), B_ / TILE_M), 256, 0, stream>>>(
        axi, C, t120, B_, NVAR, NC);                            // res @ C
    k_lamnew<<<cdiv(B_ * NVAR, 256), 256, 0, stream>>>(lam, t120, lamnew);
    gemm_f32_wmma<false><<<dim3(cdiv(NVAR, TILE_N), B_ / TILE_M), 256, 0, stream>>>(
        xi, H, ybuf, B_, NVAR, NVAR);                           // Y = xi @ H (sym)
    k_rowdot<<<B_, 256, 0, stream>>>(xi, g, NVAR, xg);
    k_qpnorm<<<B_ / 16, 256, 0, stream>>>(ybuf, xi, xg, qpres + (size_t)it * B_);
    k_concat<<<cdiv(B_ * RIN, 256), 256, 0, stream>>>(s, lam, snew, lamnew, rbuf);
    gemm_f32_wmma<true><<<dim3(cdiv(G3, TILE_N), B_ / TILE_M), 256, 0, stream>>>(
        rbuf, W_ih, gi, B_, G3, RIN);                           // r @ W_ih^T
    gemm_f32_wmma<true><<<dim3(cdiv(G3, TILE_N), B_ / TILE_M), 256, 0, stream>>>(
        h, W_hh, gh, B_, G3, HID);                              // h @ W_hh^T
    k_gates<<<cdiv(B_ * HID, 256), 256, 0, stream>>>(gi, gh, b_ih, b_hh, h);
    gemm_f32_wmma<true><<<dim3(cdiv(NC + NVAR, TILE_N), B_ / TILE_M), 256, 0, stream>>>(
        h, W_out, outb, B_, NC + NVAR, HID);                    // h_new @ W_out^T
    k_final<<<B_, 256, 0, stream>>>(outb, b_out, snew, lamnew, s, lam,
                                    fixedp + (size_t)it * B_);
  }
}